// GCNGuard_56040733278696
// MI455X (gfx1250) — compile-verified
//
#include <hip/hip_runtime.h>
#include <math.h>

// GCNGuard forward for MI455X (gfx1250, wave32).
// Dense linears use V_WMMA_F32_16X16X4_F32 (exact f32). Edge attention and
// SPMM are wave-per-edge gather/scatter kernels (L2-resident working set).

#define NNODES 50000
#define NEDGES 800000
#define SIMTH  0.1f

typedef __attribute__((ext_vector_type(2))) float v2f;
typedef __attribute__((ext_vector_type(8))) float v8f;

__global__ void zero_kernel(float* __restrict__ p, int n) {
  int i = blockIdx.x * blockDim.x + threadIdx.x;
  if (i < n) p[i] = 0.0f;
}

__global__ void relu_kernel(float* __restrict__ p, int n) {
  int i = blockIdx.x * blockDim.x + threadIdx.x;
  if (i < n) p[i] = fmaxf(p[i], 0.0f);
}

// One wave per node: ||x_i||_2 over 128 floats (lane loads float4 = 16B).
__global__ void node_norm_kernel(const float* __restrict__ X,
                                 float* __restrict__ nrm, int n_nodes) {
  int gid  = blockIdx.x * blockDim.x + threadIdx.x;
  int node = gid >> 5, lane = gid & 31;
  if (node >= n_nodes) return;
  float4 v = ((const float4*)(X + (size_t)node * 128))[lane];
  float s = v.x * v.x + v.y * v.y + v.z * v.z + v.w * v.w;
#pragma unroll
  for (int off = 16; off; off >>= 1) s += __shfl_xor(s, off, 32);
  if (lane == 0) nrm[node] = sqrtf(s);
}

// One wave per edge: thresholded cosine sim; atomic segment-sum into rowsum.
__global__ void edge_sim_kernel(const float* __restrict__ X,
                                const float* __restrict__ nrm,
                                const int* __restrict__ row,
                                const int* __restrict__ col,
                                const float* __restrict__ maskb, int use_mask,
                                float* __restrict__ sim,
                                float* __restrict__ rowsum, int n_edges) {
  int gid = blockIdx.x * blockDim.x + threadIdx.x;
  int e = gid >> 5, lane = gid & 31;
  if (e >= n_edges) return;
  int r = row[e], c = col[e];
  float4 a = ((const float4*)(X + (size_t)r * 128))[lane];
  float4 b = ((const float4*)(X + (size_t)c * 128))[lane];
  float s = a.x * b.x + a.y * b.y + a.z * b.z + a.w * b.w;
#pragma unroll
  for (int off = 16; off; off >>= 1) s += __shfl_xor(s, off, 32);
  if (lane == 0) {
    float denom = fmaxf(nrm[r] * nrm[c], 1e-12f);
    float sv = s / denom;
    float m = use_mask ? maskb[e] : 1.0f;
    sv = (m != 0.0f && sv >= SIMTH) ? sv : 0.0f;
    sim[e] = sv;
    if (sv > 0.0f) atomicAdd(&rowsum[r], sv);
  }
}

// Per edge: l1-normalize, w_edge = exp(norm) on active edges, count degree,
// persist the active mask (consumed by the layer-1 attention pass).
__global__ void edge_norm_kernel(const int* __restrict__ row,
                                 const float* __restrict__ rowsum,
                                 float* __restrict__ we,   // in: sim, out: w_edge
                                 float* __restrict__ maskb,
                                 float* __restrict__ deg, int n_edges) {
  int e = blockIdx.x * blockDim.x + threadIdx.x;
  if (e >= n_edges) return;
  int r = row[e];
  float sv = we[e];
  float rs = rowsum[r];
  float nv = (rs > 0.0f) ? sv / fmaxf(rs, 1e-30f) : 0.0f;
  bool active = nv > 0.0f;
  we[e] = active ? expf(nv) : 0.0f;
  maskb[e] = active ? 1.0f : 0.0f;
  if (active) atomicAdd(&deg[r], 1.0f);
}

// Per node: w_self = exp(1/(deg+1)); also reset rowsum/deg for the next pass.
__global__ void self_weight_kernel(float* __restrict__ deg,
                                   float* __restrict__ rowsum,
                                   float* __restrict__ wself, int n_nodes) {
  int i = blockIdx.x * blockDim.x + threadIdx.x;
  if (i >= n_nodes) return;
  wself[i] = expf(1.0f / (deg[i] + 1.0f));
  deg[i] = 0.0f;
  rowsum[i] = 0.0f;
}

// H[M][N] = sum_k X[M][k] * W[N][k] + bias[N], K = 128, via f32 WMMA.
// blockDim = 32 * (NOUT/16); wave w owns n-tile w; block owns 16 M-rows.
// A layout: lanes 0-15 -> M=lane, K={k,k+1}; lanes 16-31 -> K={k+2,k+3}.
// B layout (mirror over N): lane l -> N=l == row l of W (contiguous loads).
// D layout: c[r] -> D[M = r + 8*(lane>=16)][N = lane&15].
template <int NOUT>
__global__ void gemm_wmma_kernel(const float* __restrict__ X,
                                 const float* __restrict__ W,
                                 const float* __restrict__ bias,
                                 float* __restrict__ H) {
  const int lane  = threadIdx.x & 31;
  const int wave  = threadIdx.x >> 5;   // n-tile index
  const int half  = lane >> 4;
  const int l     = lane & 15;
  const int Mbase = blockIdx.x * 16;
  const float* arow = X + (size_t)(Mbase + l) * 128 + half * 2;
  const float* brow = W + (size_t)(wave * 16 + l) * 128 + half * 2;
  v8f c = {};
#pragma unroll
  for (int k = 0; k < 128; k += 4) {
    v2f a = *(const v2f*)(arow + k);
    v2f b = *(const v2f*)(brow + k);
    c = __builtin_amdgcn_wmma_f32_16x16x4_f32(
        /*neg_a=*/false, a, /*neg_b=*/false, b,
        /*c_mod=*/(short)0, c, /*reuse_a=*/false, /*reuse_b=*/false);
  }
  const float bv = bias[wave * 16 + l];
  float* hp = H + (size_t)(Mbase + half * 8) * NOUT + wave * 16 + l;
#pragma unroll
  for (int r = 0; r < 8; r++) hp[(size_t)r * NOUT] = c[r] + bv;
}

// out = w_self[i] * h[i][:]  (elementwise seed before edge scatter)
template <int D>
__global__ void spmm_init_kernel(const float* __restrict__ H,
                                 const float* __restrict__ wself,
                                 float* __restrict__ out, int n_nodes) {
  int idx = blockIdx.x * blockDim.x + threadIdx.x;
  if (idx >= n_nodes * D) return;
  out[idx] = wself[idx / D] * H[idx];
}

// One wave per edge: out[row] += w_e * h[col] via f32 global atomics.
template <int D>
__global__ void spmm_edge_kernel(const float* __restrict__ H,
                                 const float* __restrict__ we,
                                 const int* __restrict__ row,
                                 const int* __restrict__ col,
                                 float* __restrict__ out, int n_edges) {
  int gid = blockIdx.x * blockDim.x + threadIdx.x;
  int e = gid >> 5, lane = gid & 31;
  if (e >= n_edges) return;
  float w = we[e];                 // wave-uniform
  if (w == 0.0f) return;
  int r = row[e], c = col[e];
  const float* hc = H + (size_t)c * D;
  float* orow = out + (size_t)r * D;
  constexpr int PER = D / 32;      // 4 (D=128) or 2 (D=64)
#pragma unroll
  for (int j = 0; j < PER; j++) {
    int k = j * 32 + lane;
    atomicAdd(&orow[k], w * hc[k]);
  }
}

extern "C" void kernel_launch(void* const* d_in, const int* in_sizes, int n_in,
                              void* d_out, int out_size, void* d_ws, size_t ws_size,
                              hipStream_t stream) {
  const float* x  = (const float*)d_in[0];   // [50000,128]
  const float* W0 = (const float*)d_in[1];   // [128,128]
  const float* b0 = (const float*)d_in[2];   // [128]
  const float* W1 = (const float*)d_in[3];   // [64,128]
  const float* b1 = (const float*)d_in[4];   // [64]
  const int* erow = (const int*)d_in[5];     // [800000]
  const int* ecol = (const int*)d_in[6];     // [800000]
  float* out = (float*)d_out;                // [50000,64]

  // Workspace layout (floats): ~58.4 MB total.
  float* ws     = (float*)d_ws;
  float* rowsum = ws;                 //   50000
  float* deg    = ws + 50000;         //   50000 (contiguous with rowsum for zeroing)
  float* wself  = ws + 100000;        //   50000
  float* nrm    = ws + 150000;        //   50000
  float* wedge  = ws + 200000;        //  800000 (sim, then w_edge)
  float* maskb  = ws + 1000000;       //  800000 (layer-0 active mask)
  float* hbuf   = ws + 1800000;       // 6400000 (linear output; reused layer 1)
  float* x1     = ws + 8200000;       // 6400000 (layer-0 features)

  const int TB = 256;
  const int nodeWaveBlocks = (NNODES * 32 + TB - 1) / TB;   // wave-per-node
  const int edgeWaveBlocks = (NEDGES * 32 + TB - 1) / TB;   // wave-per-edge
  const int edgeBlocks     = (NEDGES + TB - 1) / TB;
  const int nodeBlocks     = (NNODES + TB - 1) / TB;

  // ---------------- Layer 0: attention ----------------
  zero_kernel<<<(100000 + TB - 1) / TB, TB, 0, stream>>>(rowsum, 100000);
  node_norm_kernel<<<nodeWaveBlocks, TB, 0, stream>>>(x, nrm, NNODES);
  edge_sim_kernel<<<edgeWaveBlocks, TB, 0, stream>>>(x, nrm, erow, ecol,
                                                     maskb, /*use_mask=*/0,
                                                     wedge, rowsum, NEDGES);
  edge_norm_kernel<<<edgeBlocks, TB, 0, stream>>>(erow, rowsum, wedge, maskb,
                                                  deg, NEDGES);
  self_weight_kernel<<<nodeBlocks, TB, 0, stream>>>(deg, rowsum, wself, NNODES);

  // ---------------- Layer 0: linear (WMMA) + SPMM + ReLU ----------------
  gemm_wmma_kernel<128><<<NNODES / 16, 32 * (128 / 16), 0, stream>>>(x, W0, b0, hbuf);
  spmm_init_kernel<128><<<(NNODES * 128 + TB - 1) / TB, TB, 0, stream>>>(hbuf, wself, x1, NNODES);
  spmm_edge_kernel<128><<<edgeWaveBlocks, TB, 0, stream>>>(hbuf, wedge, erow, ecol, x1, NEDGES);
  relu_kernel<<<(NNODES * 128 + TB - 1) / TB, TB, 0, stream>>>(x1, NNODES * 128);

  // ---------------- Layer 1: attention (masked by layer-0 actives) ----------------
  node_norm_kernel<<<nodeWaveBlocks, TB, 0, stream>>>(x1, nrm, NNODES);
  edge_sim_kernel<<<edgeWaveBlocks, TB, 0, stream>>>(x1, nrm, erow, ecol,
                                                     maskb, /*use_mask=*/1,
                                                     wedge, rowsum, NEDGES);
  edge_norm_kernel<<<edgeBlocks, TB, 0, stream>>>(erow, rowsum, wedge, maskb,
                                                  deg, NEDGES);
  self_weight_kernel<<<nodeBlocks, TB, 0, stream>>>(deg, rowsum, wself, NNODES);

  // ---------------- Layer 1: linear (WMMA) + SPMM (no activation) ----------------
  gemm_wmma_kernel<64><<<NNODES / 16, 32 * (64 / 16), 0, stream>>>(x1, W1, b1, hbuf);
  spmm_init_kernel<64><<<(NNODES * 64 + TB - 1) / TB, TB, 0, stream>>>(hbuf, wself, out, NNODES);
  spmm_edge_kernel<64><<<edgeWaveBlocks, TB, 0, stream>>>(hbuf, wedge, erow, ecol, out, NEDGES);
}